// DirectionalConvLayer_52553219834290
// MI455X (gfx1250) — compile-verified
//
#include <hip/hip_runtime.h>

typedef float v2f __attribute__((ext_vector_type(2)));
typedef float v8f __attribute__((ext_vector_type(8)));

#define NB    256
#define CIN   256
#define COUT  256
#define SS    225      // 15*15 spatial
#define CK    32       // C_in chunk per LDS stage
#define CKP   34       // padded k-stride (even -> aligned b64, conflict-free banks)
#define PLANE 289      // 17*17 haloed plane
#define NJ    4        // 16-wide n-tiles per wave
#define NC    2        // 16-wide cout tiles per wave

__global__ __launch_bounds__(256, 1)
void dirconv_wmma_f32(const float* __restrict__ x0, const float* __restrict__ x1,
                      const float* __restrict__ x2, const float* __restrict__ x3,
                      const float* __restrict__ wgt, const float* __restrict__ bias,
                      float* __restrict__ out)
{
    __shared__ __align__(16) float xs[PLANE * CKP];   // 39304 B, zero halo
    __shared__ __align__(16) float ws[3 * 64 * CKP];  // 26112 B

    const int tid  = threadIdx.x;
    const int lane = tid & 31;
    const int wv   = tid >> 5;            // 0..7
    const int blk  = blockIdx.x;          // 4096 blocks
    const int cog  = blk & 3;             // C_out group of 64
    const int b    = (blk >> 2) & 255;
    const int d    = blk >> 10;           // direction

    const float* xin = (d == 0) ? x0 : (d == 1) ? x1 : (d == 2) ? x2 : x3;
    // tap0 offset in the 17-wide haloed plane; taps = {dt0, 0, -dt0}
    const int dt0 = (d == 0) ? -1 : (d == 1) ? -17 : (d == 2) ? -18 : -16;

    const int cp = wv & 1;                // cout-tile pair: tiles 2cp, 2cp+1
    const int nq = wv >> 1;               // n-tile quarter: tiles 4nq..4nq+3
    const int lm = lane & 15;             // M (A) / N (B,C,D) index
    const int k2 = (lane >> 4) * 2;       // K sub-offset per lane half

    // Per-lane spatial mapping for this wave's 4 n-tiles
    int off[NJ], valid[NJ];
#pragma unroll
    for (int jj = 0; jj < NJ; ++jj) {
        int s = (nq * 4 + jj) * 16 + lm;
        int v = (s < SS);
        int sc = v ? s : 0;
        int h = sc / 15, w = sc - h * 15;
        off[jj]   = (h + 1) * 17 + (w + 1);
        valid[jj] = v;
    }

    // Zero entire xs once: halo stays zero, interior rewritten each chunk.
    for (int i = tid; i < PLANE * CKP; i += 256) xs[i] = 0.f;

    v8f acc[NC][NJ];
#pragma unroll
    for (int ci = 0; ci < NC; ++ci)
#pragma unroll
        for (int jj = 0; jj < NJ; ++jj)
            acc[ci][jj] = v8f{0.f, 0.f, 0.f, 0.f, 0.f, 0.f, 0.f, 0.f};

    const int cbase = cog * 64;
    const float* xg = xin + (size_t)b * CIN * SS;

    // Haloed-plane interior offset for this thread's staging slot
    int soff = 0;
    if (tid < SS) { int h = tid / 15, w = tid - h * 15; soff = (h + 1) * 17 + (w + 1); }

    for (int c0 = 0; c0 < CIN; c0 += CK) {
        __syncthreads();   // previous compute done before overwriting LDS

        // Stage input chunk: coalesced global, conflict-free LDS (stride CKP)
        if (tid < SS) {
            const float* xp = xg + (size_t)c0 * SS + tid;
            float* xd = xs + soff * CKP;
#pragma unroll 4
            for (int k = 0; k < CK; ++k)
                xd[k] = xp[(size_t)k * SS];
        }
        // Stage weights: ws[tap][cout64][k]  (k contiguous per lane)
        for (int i = tid; i < 3 * CK * 64; i += 256) {
            int k = i & 31, c = (i >> 5) & 63, tp = i >> 11;
            ws[tp * (64 * CKP) + c * CKP + k] =
                wgt[(size_t)(cbase + c) * (CIN * 3) + (c0 + k) * 3 + tp];
        }
        __syncthreads();

#pragma unroll
        for (int tap = 0; tap < 3; ++tap) {
            const int dtap = (tap == 0) ? dt0 : (tap == 1) ? 0 : -dt0;
            const float* ap0 = ws + tap * (64 * CKP) + ((cp * 2 + 0) * 16 + lm) * CKP + k2;
            const float* ap1 = ws + tap * (64 * CKP) + ((cp * 2 + 1) * 16 + lm) * CKP + k2;
            const float* bp[NJ];
#pragma unroll
            for (int jj = 0; jj < NJ; ++jj) {
                int o = valid[jj] ? (off[jj] + dtap) : 0;   // 0 -> zero halo corner
                bp[jj] = xs + o * CKP + k2;
            }
#pragma unroll
            for (int kk = 0; kk < CK; kk += 4) {
                v2f a0 = *(const v2f*)(ap0 + kk);           // ds_load_b64
                v2f a1 = *(const v2f*)(ap1 + kk);           // ds_load_b64
#pragma unroll
                for (int jj = 0; jj < NJ; ++jj) {
                    v2f bb = *(const v2f*)(bp[jj] + kk);    // ds_load_b64, reused 2x
                    acc[0][jj] = __builtin_amdgcn_wmma_f32_16x16x4_f32(
                        false, a0, false, bb, (short)0, acc[0][jj], false, false);
                    acc[1][jj] = __builtin_amdgcn_wmma_f32_16x16x4_f32(
                        false, a1, false, bb, (short)0, acc[1][jj], false, false);
                }
            }
        }
    }

    // Epilogue: bias + store (C/D layout: VGPR r -> M=r (lanes 0-15) / r+8 (16-31))
    const int hh8 = (lane >> 4) * 8;
    float* og = out + ((size_t)d * NB + b) * COUT * SS;
#pragma unroll
    for (int ci = 0; ci < NC; ++ci) {
#pragma unroll
        for (int jj = 0; jj < NJ; ++jj) {
            int s = (nq * 4 + jj) * 16 + lm;
            if (s < SS) {
#pragma unroll
                for (int r = 0; r < 8; ++r) {
                    int co = cbase + (cp * 2 + ci) * 16 + hh8 + r;
                    og[(size_t)co * SS + s] = acc[ci][jj][r] + bias[co];
                }
            }
        }
    }
}

extern "C" void kernel_launch(void* const* d_in, const int* in_sizes, int n_in,
                              void* d_out, int out_size, void* d_ws, size_t ws_size,
                              hipStream_t stream) {
    const float* x0 = (const float*)d_in[0];
    const float* x1 = (const float*)d_in[1];
    const float* x2 = (const float*)d_in[2];
    const float* x3 = (const float*)d_in[3];
    const float* w  = (const float*)d_in[4];
    const float* bs = (const float*)d_in[5];
    float* out = (float*)d_out;

    dim3 grid(4 * NB * 4);   // dir * batch * cout-group
    dim3 block(256);
    dirconv_wmma_f32<<<grid, block, 0, stream>>>(x0, x1, x2, x3, w, bs, out);
}